// ModulatedChunks_56367150793586
// MI455X (gfx1250) — compile-verified
//
#include <hip/hip_runtime.h>
#include <hip/hip_bf16.h>
#include <math.h>
#include <stdint.h>

// ---------------------------------------------------------------------------
// ModulatedChunks on MI455X (gfx1250).
// Key dedup: cic==4 for both configs => mode/pooled/mod/pred depend only on
// chunk start p = nw + 4k.  Two 512x512-K fp32 WMMA GEMMs dominate; the rest
// is bandwidth (gathers/scatters).  Problem is memory bound (~200MB @23.3TB/s)
// so stay fp32 and use V_WMMA_F32_16X16X4_F32.  B-tiles are staged into LDS
// with GLOBAL_LOAD_ASYNC_TO_LDS_B128 (ASYNCcnt) to overlap with A-tile VALU.
// ---------------------------------------------------------------------------

typedef __attribute__((ext_vector_type(2))) float v2f;
typedef __attribute__((ext_vector_type(8))) float v8f;
typedef int v4i_gcc __attribute__((vector_size(4 * sizeof(int))));

#if defined(__has_builtin)
#if __has_builtin(__builtin_amdgcn_wmma_f32_16x16x4_f32)
#define HAVE_WMMA_F32 1
#endif
#if __has_builtin(__builtin_amdgcn_global_load_async_to_lds_b128)
#define HAVE_ASYNC_LDS 1
#endif
#if __has_builtin(__builtin_amdgcn_s_wait_asynccnt)
#define HAVE_WAIT_ASYNC 1
#endif
#endif

#define Bb   16
#define Tt   1024
#define NQ   25
#define Cc   512
#define QD   300
#define STN  8
#define NW0  1009
#define NW1  993
#define OUT0 (Bb*2*NW0*4)      /* 129152 */
#define OUT1 (Bb*2*NW1*8)      /* 254208 */
#define OFF_ST  (OUT0+OUT1)    /* 383360 */
#define OFF_END (OFF_ST+Bb*STN)

#define LOG1E4_OVER_C 0.017988945f   /* ln(10000)/512 */
#define LOG1E4_OVER_Q 0.030701134f   /* ln(10000)/300 */

__device__ __forceinline__ unsigned long long mix64(unsigned long long z) {
  z += 0x9E3779B97F4A7C15ull;
  z = (z ^ (z >> 30)) * 0xBF58476D1CE4E5B9ull;
  z = (z ^ (z >> 27)) * 0x94D049BB133111EBull;
  return z ^ (z >> 31);
}

// async global->LDS 16-byte copy (per-lane), tracked by ASYNCcnt
__device__ __forceinline__ void async_b128(const float* g, void* lds_generic) {
#if HAVE_ASYNC_LDS
  typedef __attribute__((address_space(1))) v4i_gcc glb_v4i;
  typedef __attribute__((address_space(3))) v4i_gcc lds_v4i;
  __builtin_amdgcn_global_load_async_to_lds_b128(
      (glb_v4i*)(uintptr_t)g,
      (lds_v4i*)(uint32_t)(uintptr_t)lds_generic, 0, 0);
#else
  (void)g; (void)lds_generic;
#endif
}

__device__ __forceinline__ void wait_async0() {
#if HAVE_ASYNC_LDS
#if HAVE_WAIT_ASYNC
  __builtin_amdgcn_s_wait_asynccnt(0);
#else
  asm volatile("s_wait_asynccnt 0x0" ::: "memory");
#endif
#endif
}

// ---------------- positional encoding table for video (T x C) --------------
__global__ __launch_bounds__(256) void pe_kernel(float* __restrict__ pe) {
  int idx = blockIdx.x * 256 + threadIdx.x;
  if (idx >= Tt * Cc) return;
  int t = idx >> 9, k = idx & (Cc - 1);
  float div = __expf(-(float)(k & ~1) * LOG1E4_OVER_C);
  float ang = (float)t * div;
  pe[idx] = (k & 1) ? __cosf(ang) : __sinf(ang);
}

// ---------------- enc1/enc2 = (q+PE) @ W_seq{1,2} + b ----------------------
__global__ __launch_bounds__(256) void enc_kernel(
    const float* __restrict__ qf, const float* __restrict__ Ws1,
    const float* __restrict__ bs1, const float* __restrict__ Ws2,
    const float* __restrict__ bs2, float* __restrict__ enc1,
    float* __restrict__ enc2) {
  int idx = blockIdx.x * 256 + threadIdx.x;
  if (idx >= Bb * NQ * Cc) return;
  int n = idx & (Cc - 1);
  int rest = idx >> 9;
  int q = rest % NQ, b = rest / NQ;
  const float* qr = qf + (b * NQ + q) * QD;
  float a1 = 0.f, a2 = 0.f;
  for (int k = 0; k < QD; ++k) {
    float div = __expf(-(float)(k & ~1) * LOG1E4_OVER_Q);
    float ang = (float)q * div;
    float v = qr[k] + ((k & 1) ? __cosf(ang) : __sinf(ang));
    a1 += v * Ws1[k * Cc + n];
    a2 += v * Ws2[k * Cc + n];
  }
  enc1[idx] = a1 + bs1[n];
  enc2[idx] = a2 + bs2[n];
}

// ---------------- fp32 WMMA GEMM core (128x64 tile, 8 waves) ---------------
#define BM 128
#define BN 64
#define KC 16
#define BPAD 4  /* keeps 16B alignment for async B128 LDS writes */

__device__ __forceinline__ void wmma_tile_mac(
    const float (*As)[KC + 1], const float (*Bs)[BN + BPAD], v8f acc[4],
    int mW, int lo, int hi) {
#pragma unroll
  for (int kk = 0; kk < KC; kk += 4) {
#if HAVE_WMMA_F32
    v2f a;
    a.x = As[mW + lo][kk + 2 * hi];
    a.y = As[mW + lo][kk + 2 * hi + 1];
#pragma unroll
    for (int tB = 0; tB < 4; ++tB) {
      v2f bf;
      bf.x = Bs[kk + 2 * hi][tB * 16 + lo];
      bf.y = Bs[kk + 2 * hi + 1][tB * 16 + lo];
      acc[tB] = __builtin_amdgcn_wmma_f32_16x16x4_f32(
          false, a, false, bf, (short)0, acc[tB], false, false);
    }
#else
#pragma unroll
    for (int tB = 0; tB < 4; ++tB) {
      int colL = tB * 16 + lo;
#pragma unroll
      for (int r = 0; r < 8; ++r) {
        int rowL = mW + r + 8 * hi;
        float s = acc[tB][r];
        for (int k2 = 0; k2 < 4; ++k2) s += As[rowL][kk + k2] * Bs[kk + k2][colL];
        acc[tB][r] = s;
      }
    }
#endif
  }
}

// stage one 16x64 B tile: async DMA if available, else float4 via VGPRs
__device__ __forceinline__ void stage_B(const float* __restrict__ W, int k0,
                                        int nBlk, int br, int bc,
                                        float (*Bs)[BN + BPAD]) {
  const float* gsrc = W + (k0 + br) * Cc + nBlk + bc;
#if HAVE_ASYNC_LDS
  async_b128(gsrc, &Bs[br][bc]);
#else
  if (k0 + KC < Cc) __builtin_prefetch(W + (k0 + KC + br) * Cc + nBlk + bc, 0, 0);
  float4 w = *reinterpret_cast<const float4*>(gsrc);
  Bs[br][bc + 0] = w.x; Bs[br][bc + 1] = w.y;
  Bs[br][bc + 2] = w.z; Bs[br][bc + 3] = w.w;
#endif
}

// GEMM1: v = (vis + PE) @ W_vid1 + b_vid1
__global__ __launch_bounds__(256) void gemm1_kernel(
    const float* __restrict__ vis, const float* __restrict__ pe,
    const float* __restrict__ W, const float* __restrict__ bias,
    float* __restrict__ out) {
  __shared__ float As[BM][KC + 1];
  __shared__ float Bs[KC][BN + BPAD];
  int tid = threadIdx.x;
  int wave = tid >> 5, lane = tid & 31;
  int lo = lane & 15, hi = lane >> 4;
  int mBlk = blockIdx.x * BM, nBlk = blockIdx.y * BN;
  int mW = wave * 16;
  v8f acc[4];
#pragma unroll
  for (int i = 0; i < 4; ++i)
#pragma unroll
    for (int j = 0; j < 8; ++j) acc[i][j] = 0.f;

  int loadRow = tid >> 1;
  int loadK = (tid & 1) * 8;
  int gRow = mBlk + loadRow;
  int tTime = gRow & (Tt - 1);
  int br = tid >> 4, bc = (tid & 15) * 4;

  for (int k0 = 0; k0 < Cc; k0 += KC) {
    stage_B(W, k0, nBlk, br, bc, Bs);  // async DMA overlaps with A VALU below
    const float4* gA = reinterpret_cast<const float4*>(vis + gRow * Cc + k0 + loadK);
    const float4* gP = reinterpret_cast<const float4*>(pe + tTime * Cc + k0 + loadK);
    float4 a0 = gA[0], a1 = gA[1];
    float4 p0 = gP[0], p1 = gP[1];
    As[loadRow][loadK + 0] = a0.x + p0.x; As[loadRow][loadK + 1] = a0.y + p0.y;
    As[loadRow][loadK + 2] = a0.z + p0.z; As[loadRow][loadK + 3] = a0.w + p0.w;
    As[loadRow][loadK + 4] = a1.x + p1.x; As[loadRow][loadK + 5] = a1.y + p1.y;
    As[loadRow][loadK + 6] = a1.z + p1.z; As[loadRow][loadK + 7] = a1.w + p1.w;
    wait_async0();
    __syncthreads();
    wmma_tile_mac(As, Bs, acc, mW, lo, hi);
    __syncthreads();
  }
  int row0 = mBlk + mW + 8 * hi;
#pragma unroll
  for (int tB = 0; tB < 4; ++tB) {
    int col = nBlk + tB * 16 + lo;
    float bb = bias[col];
#pragma unroll
    for (int r = 0; r < 8; ++r) out[(row0 + r) * Cc + col] = acc[tB][r] + bb;
  }
}

// ---------------- per-(b,t): sim -> probs -> logits -> gumbel argmax -------
__global__ __launch_bounds__(256) void sample_kernel(
    const float* __restrict__ v, const float* __restrict__ enc1,
    int* __restrict__ lab0, int* __restrict__ lab1) {
  int idx = blockIdx.x * 256 + threadIdx.x;
  if (idx >= Bb * Tt) return;
  int b = idx >> 10;
  float acc[NQ];
#pragma unroll
  for (int q = 0; q < NQ; ++q) acc[q] = 0.f;
  const float* vr = v + idx * Cc;
  const float* e = enc1 + b * NQ * Cc;
  for (int k = 0; k < Cc; ++k) {
    float vv = vr[k];
#pragma unroll
    for (int q = 0; q < NQ; ++q) acc[q] += vv * e[q * Cc + k];
  }
  float mn = acc[0];
#pragma unroll
  for (int q = 1; q < NQ; ++q) mn = fminf(mn, acc[q]);
  float sum = 0.f;
#pragma unroll
  for (int q = 0; q < NQ; ++q) sum += acc[q] - mn;
  float inv = 1.f / sum;
  for (int jj = 0; jj < 2; ++jj) {
    int best = 0;
    float bestv = -3.4e38f;
#pragma unroll
    for (int q = 0; q < NQ; ++q) {
      float p = (acc[q] - mn) * inv;
      float lg = __logf(fmaxf(p, 1e-30f));
      unsigned long long seed =
          ((unsigned long long)(unsigned)(idx * 2 + jj) << 32) ^
          (unsigned long long)(unsigned)(q * 0x9E3779B9u + 0x85EBCA6Bu);
      unsigned long long h = mix64(seed);
      float u = ((float)(h >> 40) + 0.5f) * (1.0f / 16777216.0f);
      float g = -__logf(-__logf(u));
      float val = lg + g;
      if (val > bestv) { bestv = val; best = q; }
    }
    (jj ? lab1 : lab0)[idx] = best;
  }
}

// ---------------- windowed bincount-mode over 4 consecutive labels ---------
__global__ __launch_bounds__(256) void mode_kernel(
    const int* __restrict__ lab0, const int* __restrict__ lab1,
    int* __restrict__ mode0, int* __restrict__ mode1) {
  int idx = blockIdx.x * 256 + threadIdx.x;
  if (idx >= Bb * Tt) return;
  int b = idx >> 10, p = idx & (Tt - 1);
  for (int jj = 0; jj < 2; ++jj) {
    const int* lab = jj ? lab1 : lab0;
    int l[4];
#pragma unroll
    for (int j = 0; j < 4; ++j) l[j] = lab[b * Tt + min(p + j, Tt - 1)];
    int best = 0x7fffffff, bc = 0;
#pragma unroll
    for (int j = 0; j < 4; ++j) {
      int c = 0;
#pragma unroll
      for (int j2 = 0; j2 < 4; ++j2) c += (l[j2] == l[j]);
      if (c > bc || (c == bc && l[j] < best)) { bc = c; best = l[j]; }
    }
    (jj ? mode1 : mode0)[idx] = best;
  }
}

// GEMM2: pv = avg4(v) @ W_vid2 + b_vid2 ; modp{0,1} = pv * enc2[mode{0,1}]
__global__ __launch_bounds__(256) void gemm2_kernel(
    const float* __restrict__ v, const float* __restrict__ W,
    const float* __restrict__ bias, const float* __restrict__ enc2,
    const int* __restrict__ mode0, const int* __restrict__ mode1,
    float* __restrict__ modp0, float* __restrict__ modp1) {
  __shared__ float As[BM][KC + 1];
  __shared__ float Bs[KC][BN + BPAD];
  int tid = threadIdx.x;
  int wave = tid >> 5, lane = tid & 31;
  int lo = lane & 15, hi = lane >> 4;
  int mBlk = blockIdx.x * BM, nBlk = blockIdx.y * BN;
  int mW = wave * 16;
  v8f acc[4];
#pragma unroll
  for (int i = 0; i < 4; ++i)
#pragma unroll
    for (int j = 0; j < 8; ++j) acc[i][j] = 0.f;

  int loadRow = tid >> 1;
  int loadK = (tid & 1) * 8;
  int gRow = mBlk + loadRow;
  int gb = gRow >> 10, gp = gRow & (Tt - 1);
  int br = tid >> 4, bc = (tid & 15) * 4;

  for (int k0 = 0; k0 < Cc; k0 += KC) {
    stage_B(W, k0, nBlk, br, bc, Bs);  // async DMA overlaps with pooling below
    float s[8];
#pragma unroll
    for (int i = 0; i < 8; ++i) s[i] = 0.f;
#pragma unroll
    for (int j = 0; j < 4; ++j) {
      int r = gb * Tt + min(gp + j, Tt - 1);
      const float4* g = reinterpret_cast<const float4*>(v + r * Cc + k0 + loadK);
      float4 x0 = g[0], x1 = g[1];
      s[0] += x0.x; s[1] += x0.y; s[2] += x0.z; s[3] += x0.w;
      s[4] += x1.x; s[5] += x1.y; s[6] += x1.z; s[7] += x1.w;
    }
#pragma unroll
    for (int i = 0; i < 8; ++i) As[loadRow][loadK + i] = 0.25f * s[i];
    wait_async0();
    __syncthreads();
    wmma_tile_mac(As, Bs, acc, mW, lo, hi);
    __syncthreads();
  }
  int row0 = mBlk + mW + 8 * hi;
  int m0a[8], m1a[8];
#pragma unroll
  for (int r = 0; r < 8; ++r) {
    m0a[r] = min(max(mode0[row0 + r], 0), NQ - 1);
    m1a[r] = min(max(mode1[row0 + r], 0), NQ - 1);
  }
#pragma unroll
  for (int tB = 0; tB < 4; ++tB) {
    int col = nBlk + tB * 16 + lo;
    float bb = bias[col];
#pragma unroll
    for (int r = 0; r < 8; ++r) {
      int row = row0 + r;
      int b = row >> 10;
      float pv = acc[tB][r] + bb;
      modp0[row * Cc + col] = pv * enc2[(b * NQ + m0a[r]) * Cc + col];
      modp1[row * Cc + col] = pv * enc2[(b * NQ + m1a[r]) * Cc + col];
    }
  }
}

// ---------------- op{0,1}[b,p,c] = modp{0,1}[b,p,:] @ W_pred + b_pred ------
__global__ __launch_bounds__(256) void op_kernel(
    const float* __restrict__ modp0, const float* __restrict__ modp1,
    const float* __restrict__ Wp, const float* __restrict__ bp,
    float* __restrict__ op0, float* __restrict__ op1) {
  int idx = blockIdx.x * 256 + threadIdx.x;
  if (idx >= Bb * Tt) return;
  const float* m0 = modp0 + idx * Cc;
  const float* m1 = modp1 + idx * Cc;
  float c00 = 0.f, c01 = 0.f, c10 = 0.f, c11 = 0.f;
  for (int k = 0; k < Cc; ++k) {
    float w0 = Wp[k * 2], w1 = Wp[k * 2 + 1];
    float x0 = m0[k], x1 = m1[k];
    c00 += x0 * w0; c01 += x0 * w1;
    c10 += x1 * w0; c11 += x1 * w1;
  }
  op0[idx * 2 + 0] = c00 + bp[0]; op0[idx * 2 + 1] = c01 + bp[1];
  op1[idx * 2 + 0] = c10 + bp[0]; op1[idx * 2 + 1] = c11 + bp[1];
}

// ---------------- scatter per-position preds into [B,2,NW,NC] outputs ------
__global__ __launch_bounds__(256) void scatter_kernel(
    const float* __restrict__ op0, const float* __restrict__ op1,
    float* __restrict__ out) {
  int idx = blockIdx.x * 256 + threadIdx.x;
  if (idx >= OUT0 + OUT1) return;
  if (idx < OUT0) {
    int k = idx & 3;
    int rest = idx >> 2;
    int nw = rest % NW0; rest /= NW0;
    int c = rest & 1, b = rest >> 1;
    out[idx] = op0[(b * Tt + nw + 4 * k) * 2 + c];
  } else {
    int i2 = idx - OUT0;
    int k = i2 & 7;
    int rest = i2 >> 3;
    int nw = rest % NW1; rest /= NW1;
    int c = rest & 1, b = rest >> 1;
    out[idx] = op1[(b * Tt + nw + 4 * k) * 2 + c];
  }
}

// ---------------- AdaptiveMaxPool1d over valid prefix of flattened mod -----
__global__ __launch_bounds__(256) void maxpool_kernel(
    const float* __restrict__ modp0, const int* __restrict__ vid_len,
    float* __restrict__ mp) {
  int idx = blockIdx.x * 256 + threadIdx.x;
  if (idx >= Bb * Cc * STN) return;
  int i = idx & 7;
  int rest = idx >> 3;
  int c = rest & (Cc - 1), b = rest >> 9;
  int Lv = vid_len[b] * 4;
  int s = (i * Lv) / STN;
  int e = ((i + 1) * Lv + STN - 1) / STN;
  float mx = -3.4e38f;
  for (int l = s; l < e; ++l) {
    int p = (l >> 2) + 4 * (l & 3);  // flat (nw,k) -> position
    mx = fmaxf(mx, modp0[(b * Tt + p) * Cc + c]);
  }
  mp[idx] = mx;  // layout [B, C*8]
}

// ---------------- st/end heads: mp @ W_{st,end} + b ------------------------
__global__ __launch_bounds__(256) void stend_kernel(
    const float* __restrict__ mp, const float* __restrict__ Wst,
    const float* __restrict__ bst, const float* __restrict__ Wend,
    const float* __restrict__ bend, float* __restrict__ out) {
  int idx = threadIdx.x;  // 256 = 2*B*8
  if (idx >= 2 * Bb * STN) return;
  int w = idx / (Bb * STN);
  int r = idx % (Bb * STN);
  int b = r >> 3, i = r & 7;
  const float* Wm = w ? Wend : Wst;
  float s = (w ? bend : bst)[i];
  const float* mpb = mp + b * Cc * STN;
  for (int k = 0; k < Cc * STN; ++k) s += mpb[k] * Wm[k * STN + i];
  out[(w ? OFF_END : OFF_ST) + b * STN + i] = s;
}

// ---------------------------------------------------------------------------
extern "C" void kernel_launch(void* const* d_in, const int* in_sizes, int n_in,
                              void* d_out, int out_size, void* d_ws,
                              size_t ws_size, hipStream_t stream) {
  const float* vis    = (const float*)d_in[0];
  const float* qf     = (const float*)d_in[1];
  const int*   vlen   = (const int*)d_in[2];
  const float* W_vid1 = (const float*)d_in[3];
  const float* b_vid1 = (const float*)d_in[4];
  const float* W_vid2 = (const float*)d_in[5];
  const float* b_vid2 = (const float*)d_in[6];
  const float* W_seq1 = (const float*)d_in[7];
  const float* b_seq1 = (const float*)d_in[8];
  const float* W_seq2 = (const float*)d_in[9];
  const float* b_seq2 = (const float*)d_in[10];
  const float* W_pred = (const float*)d_in[11];
  const float* b_pred = (const float*)d_in[12];
  const float* W_st   = (const float*)d_in[13];
  const float* b_st   = (const float*)d_in[14];
  const float* W_end  = (const float*)d_in[15];
  const float* b_end  = (const float*)d_in[16];
  float* out = (float*)d_out;
  float* ws  = (float*)d_ws;

  // workspace layout (floats)
  float* pe    = ws;                       // T*C
  float* enc1  = pe + Tt * Cc;             // B*NQ*C
  float* enc2  = enc1 + Bb * NQ * Cc;      // B*NQ*C
  float* v     = enc2 + Bb * NQ * Cc;      // B*T*C
  float* modp0 = v + Bb * Tt * Cc;         // B*T*C
  float* modp1 = modp0 + Bb * Tt * Cc;     // B*T*C
  float* op0   = modp1 + Bb * Tt * Cc;     // B*T*2
  float* op1   = op0 + Bb * Tt * 2;        // B*T*2
  float* mp    = op1 + Bb * Tt * 2;        // B*C*8
  int* lab0  = (int*)(mp + Bb * Cc * STN); // B*T each
  int* lab1  = lab0 + Bb * Tt;
  int* mode0 = lab1 + Bb * Tt;
  int* mode1 = mode0 + Bb * Tt;

  pe_kernel<<<(Tt * Cc) / 256, 256, 0, stream>>>(pe);
  enc_kernel<<<(Bb * NQ * Cc) / 256, 256, 0, stream>>>(qf, W_seq1, b_seq1,
                                                       W_seq2, b_seq2, enc1, enc2);
  gemm1_kernel<<<dim3((Bb * Tt) / BM, Cc / BN), 256, 0, stream>>>(
      vis, pe, W_vid1, b_vid1, v);
  sample_kernel<<<(Bb * Tt) / 256, 256, 0, stream>>>(v, enc1, lab0, lab1);
  mode_kernel<<<(Bb * Tt) / 256, 256, 0, stream>>>(lab0, lab1, mode0, mode1);
  gemm2_kernel<<<dim3((Bb * Tt) / BM, Cc / BN), 256, 0, stream>>>(
      v, W_vid2, b_vid2, enc2, mode0, mode1, modp0, modp1);
  op_kernel<<<(Bb * Tt) / 256, 256, 0, stream>>>(modp0, modp1, W_pred, b_pred,
                                                 op0, op1);
  scatter_kernel<<<(OUT0 + OUT1 + 255) / 256, 256, 0, stream>>>(op0, op1, out);
  maxpool_kernel<<<(Bb * Cc * STN) / 256, 256, 0, stream>>>(modp0, vlen, mp);
  stend_kernel<<<1, 256, 0, stream>>>(mp, W_st, b_st, W_end, b_end, out);
}